// HeteroRGCN_64501818851543
// MI455X (gfx1250) — compile-verified
//
#include <hip/hip_runtime.h>
#include <math.h>

typedef float v2f __attribute__((ext_vector_type(2)));
typedef float v8f __attribute__((ext_vector_type(8)));

#define D_F 128
#define SLOPE 0.01f

// ---------------------------------------------------------------- fill
__global__ void hg_fill(float* __restrict__ p, float v, int n) {
    int i = blockIdx.x * blockDim.x + threadIdx.x;
    int stride = gridDim.x * blockDim.x;
    for (; i < n; i += stride) p[i] = v;
}

// ---------------------------------------------------------------- GEMM + bias via fp32 WMMA
// C[M,128] = A[M,128] @ W[128,128] + bias[128]
// block = 256 threads (8 waves), each wave owns one 16-wide N tile; grid = ceil(M/16)
__global__ void hg_gemm_bias_wmma(const float* __restrict__ A,
                                  const float* __restrict__ W,
                                  const float* __restrict__ bias,
                                  float* __restrict__ C, int M) {
    __shared__ float As[16 * 129];               // padded: conflict-free column reads
    const int tile_m = blockIdx.x * 16;
    for (int idx = threadIdx.x; idx < 16 * D_F; idx += blockDim.x) {
        int r = idx >> 7, c = idx & 127;
        int gr = tile_m + r; if (gr >= M) gr = M - 1;          // clamp (no EXEC divergence)
        As[r * 129 + c] = A[(size_t)gr * D_F + c];
    }
    __syncthreads();

    const int wave = threadIdx.x >> 5;
    const int lane = threadIdx.x & 31;
    const int n0   = wave * 16;
    const int ml   = lane & 15;                  // row-in-tile for A, col-in-tile for B
    const int kb   = (lane < 16) ? 0 : 2;        // K sub-offset per half-wave
    const float* arow = As + ml * 129;

    v8f acc = {};
    #pragma unroll
    for (int k0 = 0; k0 < D_F; k0 += 4) {
        v2f a, b;
        a.x = arow[k0 + kb];
        a.y = arow[k0 + kb + 1];
        b.x = W[(k0 + kb) * D_F + n0 + ml];
        b.y = W[(k0 + kb + 1) * D_F + n0 + ml];
        acc = __builtin_amdgcn_wmma_f32_16x16x4_f32(
            /*neg_a=*/false, a, /*neg_b=*/false, b,
            /*c_mod=*/(short)0, acc, /*reuse_a=*/false, /*reuse_b=*/false);
    }

    const int ro = (lane < 16) ? 0 : 8;
    const float bb = bias[n0 + ml];
    #pragma unroll
    for (int r = 0; r < 8; ++r) {
        int gr = tile_m + r + ro;
        if (gr < M) C[(size_t)gr * D_F + n0 + ml] = acc[r] + bb;
    }
}

// ---------------------------------------------------------------- per-node attention scalar
// out[n] = dot(Wh[n, 0:128], avec[0:128]); one wave per node, float4 per lane
__global__ void hg_node_score(const float* __restrict__ Wh,
                              const float* __restrict__ avec,
                              float* __restrict__ out, int n) {
    const int wave = threadIdx.x >> 5;
    const int lane = threadIdx.x & 31;
    const int node = blockIdx.x * 8 + wave;
    if (node >= n) return;
    float4 r  = ((const float4*)(Wh + (size_t)node * D_F))[lane];
    float4 av = ((const float4*)avec)[lane];
    float s = r.x * av.x + r.y * av.y + r.z * av.z + r.w * av.w;
    #pragma unroll
    for (int o = 16; o > 0; o >>= 1) s += __shfl_down(s, o, 32);
    if (lane == 0) out[node] = s;
}

// ---------------------------------------------------------------- edge phase 1: logits + segment max
__global__ void hg_edge_logits(const int* __restrict__ src, const int* __restrict__ dst,
                               const float* __restrict__ s_src, const float* __restrict__ s_dst,
                               float* __restrict__ e_out, float* __restrict__ emax, int E) {
    int i = blockIdx.x * blockDim.x + threadIdx.x;
    if (i >= E) return;
    float e = s_src[src[i]] + s_dst[dst[i]];
    e = (e >= 0.0f) ? e : SLOPE * e;             // leaky_relu
    e_out[i] = e;
    atomicMax(emax + dst[i], e);                 // global_atomic_max_num_f32
}

// ---------------------------------------------------------------- edge phase 2: exp + segment sum
__global__ void hg_edge_exp(const int* __restrict__ dst,
                            float* __restrict__ e_inout,
                            const float* __restrict__ emax,
                            float* __restrict__ denom, int E) {
    int i = blockIdx.x * blockDim.x + threadIdx.x;
    if (i >= E) return;
    float ex = __expf(e_inout[i] - emax[dst[i]]);
    e_inout[i] = ex;
    atomicAdd(denom + dst[i], ex);
}

// ---------------------------------------------------------------- edge phase 3: weighted scatter-sum
// one wave per edge; lane handles 4 consecutive channels
__global__ void hg_edge_agg(const int* __restrict__ src, const int* __restrict__ dst,
                            const float* __restrict__ ex, const float* __restrict__ denom,
                            const float* __restrict__ Wh_src, float* __restrict__ out, int E) {
    const int wave = threadIdx.x >> 5;
    const int lane = threadIdx.x & 31;
    const int edge = blockIdx.x * 8 + wave;
    if (edge >= E) return;
    const int s = src[edge], d = dst[edge];
    const float alpha = ex[edge] / denom[d];
    float4 m = ((const float4*)(Wh_src + (size_t)s * D_F))[lane];
    float* o = out + (size_t)d * D_F + lane * 4;
    atomicAdd(o + 0, alpha * m.x);
    atomicAdd(o + 1, alpha * m.y);
    atomicAdd(o + 2, alpha * m.z);
    atomicAdd(o + 3, alpha * m.w);
}

// ---------------------------------------------------------------- launcher
extern "C" void kernel_launch(void* const* d_in, const int* in_sizes, int n_in,
                              void* d_out, int out_size, void* d_ws, size_t ws_size,
                              hipStream_t stream) {
    const float* feat_drug = (const float*)d_in[0];
    const float* feat_dis  = (const float*)d_in[1];
    const int*   src_ind   = (const int*)d_in[2];
    const int*   dst_ind   = (const int*)d_in[3];
    const int*   src_rev   = (const int*)d_in[4];
    const int*   dst_rev   = (const int*)d_in[5];
    const int*   src_dd    = (const int*)d_in[6];
    const int*   dst_dd    = (const int*)d_in[7];
    const float* W_ind     = (const float*)d_in[8];
    const float* b_ind     = (const float*)d_in[9];
    const float* W_rev     = (const float*)d_in[10];
    const float* b_rev     = (const float*)d_in[11];
    const float* W_dd      = (const float*)d_in[12];
    const float* b_dd      = (const float*)d_in[13];
    const float* a_ind     = (const float*)d_in[14];
    const float* a_rev     = (const float*)d_in[15];
    const float* a_dd      = (const float*)d_in[16];

    const int ND = in_sizes[0] / D_F;   // 20000 drugs
    const int NS = in_sizes[1] / D_F;   // 40000 diseases
    const int E  = in_sizes[2];         // 500000 edges per etype

    // workspace layout (floats)
    float* ws = (float*)d_ws;
    float* Wh_ind = ws;  ws += (size_t)ND * D_F;
    float* Wh_rev = ws;  ws += (size_t)NS * D_F;
    float* Wh_dd  = ws;  ws += (size_t)NS * D_F;
    float* s_ind_src = ws; ws += ND;    // drug-side scalar, etype ind
    float* s_ind_dst = ws; ws += NS;
    float* s_dd_src  = ws; ws += NS;
    float* s_dd_dst  = ws; ws += NS;
    float* s_rev_src = ws; ws += NS;
    float* s_rev_dst = ws; ws += ND;
    float* emax_blk  = ws;              // emax_ind | emax_dd | emax_rev (contiguous)
    float* emax_ind  = ws; ws += NS;
    float* emax_dd   = ws; ws += NS;
    float* emax_rev  = ws; ws += ND;
    float* den_blk   = ws;              // den_ind | den_dd | den_rev (contiguous)
    float* den_ind   = ws; ws += NS;
    float* den_dd    = ws; ws += NS;
    float* den_rev   = ws; ws += ND;
    float* ex_ind    = ws; ws += E;
    float* ex_dd     = ws; ws += E;
    float* ex_rev    = ws; ws += E;

    float* h_drug = (float*)d_out;                       // [ND,128] first in tuple order
    float* h_dis  = (float*)d_out + (size_t)ND * D_F;    // [NS,128]

    // init: zero outputs & denominators, -inf segment maxes
    hg_fill<<<1024, 256, 0, stream>>>((float*)d_out, 0.0f, out_size);
    hg_fill<<<256, 256, 0, stream>>>(emax_blk, -1e30f, 2 * NS + ND);
    hg_fill<<<256, 256, 0, stream>>>(den_blk, 0.0f, 2 * NS + ND);

    // per-etype linear projections (WMMA fp32)
    hg_gemm_bias_wmma<<<(ND + 15) / 16, 256, 0, stream>>>(feat_drug, W_ind, b_ind, Wh_ind, ND);
    hg_gemm_bias_wmma<<<(NS + 15) / 16, 256, 0, stream>>>(feat_dis,  W_rev, b_rev, Wh_rev, NS);
    hg_gemm_bias_wmma<<<(NS + 15) / 16, 256, 0, stream>>>(feat_dis,  W_dd,  b_dd,  Wh_dd,  NS);

    // per-node attention scalars: s_src = Wh·a[:128], s_dst = Wh·a[128:]
    hg_node_score<<<(ND + 7) / 8, 256, 0, stream>>>(Wh_ind, a_ind,        s_ind_src, ND);
    hg_node_score<<<(NS + 7) / 8, 256, 0, stream>>>(Wh_rev, a_ind + D_F,  s_ind_dst, NS);
    hg_node_score<<<(NS + 7) / 8, 256, 0, stream>>>(Wh_dd,  a_dd,         s_dd_src,  NS);
    hg_node_score<<<(NS + 7) / 8, 256, 0, stream>>>(Wh_dd,  a_dd + D_F,   s_dd_dst,  NS);
    hg_node_score<<<(NS + 7) / 8, 256, 0, stream>>>(Wh_rev, a_rev,        s_rev_src, NS);
    hg_node_score<<<(ND + 7) / 8, 256, 0, stream>>>(Wh_ind, a_rev + D_F,  s_rev_dst, ND);

    const int eb  = (E + 255) / 256;   // 1 thread / edge
    const int ebw = (E + 7) / 8;       // 1 wave / edge

    // etype 'indication': drug -> disease
    hg_edge_logits<<<eb, 256, 0, stream>>>(src_ind, dst_ind, s_ind_src, s_ind_dst, ex_ind, emax_ind, E);
    hg_edge_exp   <<<eb, 256, 0, stream>>>(dst_ind, ex_ind, emax_ind, den_ind, E);
    hg_edge_agg   <<<ebw, 256, 0, stream>>>(src_ind, dst_ind, ex_ind, den_ind, Wh_ind, h_dis, E);

    // etype 'dd': disease -> disease (cross-etype sum into h_dis)
    hg_edge_logits<<<eb, 256, 0, stream>>>(src_dd, dst_dd, s_dd_src, s_dd_dst, ex_dd, emax_dd, E);
    hg_edge_exp   <<<eb, 256, 0, stream>>>(dst_dd, ex_dd, emax_dd, den_dd, E);
    hg_edge_agg   <<<ebw, 256, 0, stream>>>(src_dd, dst_dd, ex_dd, den_dd, Wh_dd, h_dis, E);

    // etype 'rev_indication': disease -> drug
    hg_edge_logits<<<eb, 256, 0, stream>>>(src_rev, dst_rev, s_rev_src, s_rev_dst, ex_rev, emax_rev, E);
    hg_edge_exp   <<<eb, 256, 0, stream>>>(dst_rev, ex_rev, emax_rev, den_rev, E);
    hg_edge_agg   <<<ebw, 256, 0, stream>>>(src_rev, dst_rev, ex_rev, den_rev, Wh_rev, h_drug, E);
}